// PF_PLRNN_28166395527567
// MI455X (gfx1250) — compile-verified
//
#include <hip/hip_runtime.h>
#include <cmath>

// PF_PLRNN scan: z_{t+1} = z*diag(A) + relu(z)@W_off^T + u@C^T + nz*q ; x = z@B^T + nx*r
// One wave32 per 16-sim tile; T-loop in-kernel; all matmuls via V_WMMA_F32_16X16X4_F32;
// weights resident in VGPRs; noise/input streams staged into LDS by the Tensor Data Mover
// (double-buffered, one step ahead), synchronized with s_wait_tensorcnt.

#define DIMZ   64
#define DIMX   32
#define DIMS   16
#define TSTEPS 1024
#define NSIMS  1024

typedef float v2f __attribute__((ext_vector_type(2)));
typedef float v8f __attribute__((ext_vector_type(8)));

#if __has_builtin(__builtin_amdgcn_tensor_load_to_lds)
  #define HAVE_TDM 1
  #if __clang_major__ >= 23
    #define TDM_SIX_ARG 1
  #endif
#endif

typedef unsigned int u32x4 __attribute__((ext_vector_type(4)));
typedef int          i32x4 __attribute__((ext_vector_type(4)));
typedef int          i32x8 __attribute__((ext_vector_type(8)));

// ---------------- LDS layout (floats) ----------------
// Weight B-operand fragments (staged once, then copied to VGPRs):
#define WOFF 0                   // W_off^T : 64 frags * 64 floats
#define DOFF (WOFF + 64 * 64)    // diag    : 16 frags
#define COFF (DOFF + 16 * 64)    // C^T     : 16 frags
#define BOFF (COFF + 16 * 64)    // B^T     : 32 frags
#define ZOFF (BOFF + 32 * 64)    // z relayout buffer 16x64
#define NZ0  (ZOFF + 16 * 64)    // staged noise_z, buffer 0 (16x64)
#define NZ1  (NZ0 + 1024)        // buffer 1
#define NX0  (NZ1 + 1024)        // staged noise_x (16x32)
#define NX1  (NX0 + 512)
#define UOF0 (NX1 + 512)         // staged u (16x16)
#define UOF1 (UOF0 + 256)
#define LDSZ (UOF1 + 256)        // 12800 floats = 50 KB

#ifdef HAVE_TDM
// 1-D TDM load: nElems consecutive f32 from `src` into LDS byte offset `ldsOff`.
// D# per CDNA5 ISA: group0 = {count=1 | lds_addr | global_addr | type=2},
// group1 = {data_size=4B, tensor_dim0=tile_dim0=nElems, tensor_dim1=tile_dim1=1}.
__device__ __forceinline__ void tdm_load_1d(unsigned int ldsOff, const float* src, int nElems) {
    unsigned long long a = (unsigned long long)(size_t)src;
    u32x4 g0 = { 1u,                                   // count=1, user mode
                 ldsOff,                               // lds_addr (bytes)
                 (unsigned int)a,                      // global_addr[31:0]
                 ((unsigned int)(a >> 32) & 0x01ffffffu) | 0x80000000u }; // addr[56:32] | type=2
    i32x8 g1 = { 0x00020000,                           // data_size=2 (4 bytes), no multicast
                 (nElems & 0xffff) << 16,              // tensor_dim0[15:0]
                 ((nElems >> 16) & 0xffff) | (1 << 16),// tensor_dim0[31:16] | tensor_dim1[15:0]=1
                 (nElems & 0xffff) << 16,              // tensor_dim1[31:16]=0 | tile_dim0
                 1,                                    // tile_dim1=1, tile_dim2=0
                 nElems,                               // tensor_dim0_stride (unused for 1 row)
                 0, 0 };
    i32x4 z4 = { 0, 0, 0, 0 };
#ifdef TDM_SIX_ARG
    i32x8 z8 = { 0, 0, 0, 0, 0, 0, 0, 0 };
    __builtin_amdgcn_tensor_load_to_lds(g0, g1, z4, z4, z8, 0);
#else
    __builtin_amdgcn_tensor_load_to_lds(g0, g1, z4, z4, 0);
#endif
}
#endif

__global__ __launch_bounds__(32)
void plrnn_scan_kernel(const float* __restrict__ u,
                       const float* __restrict__ z0,
                       const float* __restrict__ nzg,
                       const float* __restrict__ nxg,
                       const float* __restrict__ AW,
                       const float* __restrict__ Cm,
                       const float* __restrict__ Bm,
                       const float* __restrict__ Q,
                       const float* __restrict__ R,
                       float* __restrict__ zout,
                       float* __restrict__ xout)
{
    __shared__ float lds[LDSZ];

    const int lane = threadIdx.x;       // 0..31
    const int lo   = lane & 15;
    const int hi   = lane >> 4;         // lane-half
    const int simBase = blockIdx.x * 16;

    // ---------------- Stage invariant weights into LDS as B-operand fragments --------------
    // Fragment layout (symmetric to documented A layout for 16x16x4 f32):
    //   comp v, lane l -> B[k = 4*kk + 2*(l/16) + v][n = 16*j + (l%16)]
    for (int e = lane; e < 64 * 64; e += 32) {          // W_off^T (zero diagonal)
        int v = e & 1, l = (e >> 1) & 31, f = e >> 6;
        int kk = f >> 2, j = f & 3;
        int flo = l & 15, fhi = l >> 4;
        int k = 4 * kk + 2 * fhi + v;
        int n = 16 * j + flo;
        lds[WOFF + e] = (n == k) ? 0.0f : AW[n * DIMZ + k];
    }
    for (int e = lane; e < 16 * 64; e += 32) {          // diag(A): fragIdx = dk*4+j, kk = 4*j+dk
        int v = e & 1, l = (e >> 1) & 31, f = e >> 6;
        int dk = f >> 2, j = f & 3;
        int flo = l & 15, fhi = l >> 4;
        int n = 16 * j + flo;
        lds[DOFF + e] = (flo == 4 * dk + 2 * fhi + v) ? AW[n * DIMZ + n] : 0.0f;
    }
    for (int e = lane; e < 16 * 64; e += 32) {          // C^T (K=16)
        int v = e & 1, l = (e >> 1) & 31, f = e >> 6;
        int kk = f >> 2, j = f & 3;
        int flo = l & 15, fhi = l >> 4;
        int k = 4 * kk + 2 * fhi + v;
        int n = 16 * j + flo;
        lds[COFF + e] = Cm[n * DIMS + k];
    }
    for (int e = lane; e < 32 * 64; e += 32) {          // B^T (K=64, N=32): fragIdx = kk*2+jx
        int v = e & 1, l = (e >> 1) & 31, f = e >> 6;
        int kk = f >> 1, jx = f & 1;
        int flo = l & 15, fhi = l >> 4;
        int k = 4 * kk + 2 * fhi + v;
        int n = 16 * jx + flo;
        lds[BOFF + e] = Bm[n * DIMZ + k];
    }
    __syncthreads();

    // ---- copy all weight fragments into VGPRs (persist for whole scan; ~256 VGPRs) ----
    v2f wfr[64], dfr[16], cfr[16], bfr[32];
#pragma unroll
    for (int f = 0; f < 64; ++f) wfr[f] = *(const v2f*)&lds[WOFF + f * 64 + lane * 2];
#pragma unroll
    for (int f = 0; f < 16; ++f) dfr[f] = *(const v2f*)&lds[DOFF + f * 64 + lane * 2];
#pragma unroll
    for (int f = 0; f < 16; ++f) cfr[f] = *(const v2f*)&lds[COFF + f * 64 + lane * 2];
#pragma unroll
    for (int f = 0; f < 32; ++f) bfr[f] = *(const v2f*)&lds[BOFF + f * 64 + lane * 2];

    // Per-lane noise std: q = softplus(Q)+eps (depends only on lane%16 and N-tile)
    float qv[4], rv[2];
#pragma unroll
    for (int j = 0; j < 4; ++j) qv[j] = logf(1.0f + expf(Q[16 * j + lo])) + 1e-6f;
#pragma unroll
    for (int j = 0; j < 2; ++j) rv[j] = logf(1.0f + expf(R[16 * j + lo])) + 1e-6f;

    // ---- initial state: z0 in C/D layout (coalesced), relayout through LDS to A layout ----
    // C/D: comp rr, lane l -> z[sim = rr + 8*(l/16)][dim = 16*j + l%16]
    v8f zc[4];
#pragma unroll
    for (int j = 0; j < 4; ++j)
#pragma unroll
        for (int rr = 0; rr < 8; ++rr)
            zc[j][rr] = z0[(size_t)(simBase + rr + 8 * hi) * DIMZ + 16 * j + lo];
#pragma unroll
    for (int j = 0; j < 4; ++j)
#pragma unroll
        for (int rr = 0; rr < 8; ++rr)
            lds[ZOFF + (rr + 8 * hi) * DIMZ + 16 * j + lo] = zc[j][rr];
    asm volatile("s_wait_dscnt 0" ::: "memory");
    // A layout: za[kk], comp v, lane l -> z[sim = l%16][dim = 4*kk + 2*(l/16) + v]
    v2f za[16];
#pragma unroll
    for (int kk = 0; kk < 16; ++kk)
        za[kk] = *(const v2f*)&lds[ZOFF + lo * DIMZ + 4 * kk + 2 * hi];

#ifdef HAVE_TDM
    // prefetch streams for t=0 into buffer 0
    tdm_load_1d((unsigned)(NZ0 * 4),  &nzg[(size_t)simBase * DIMZ], 16 * DIMZ);
    tdm_load_1d((unsigned)(NX0 * 4),  &nxg[(size_t)simBase * DIMX], 16 * DIMX);
    tdm_load_1d((unsigned)(UOF0 * 4), &u  [(size_t)simBase * DIMS], 16 * DIMS);
#endif

#pragma unroll 1
    for (int t = 0; t < TSTEPS; ++t) {
        const size_t rowZ = (size_t)t * NSIMS + simBase;

        // ---- emit z_t from C/D copy (half-wave-contiguous b32 stores) ----
#pragma unroll
        for (int j = 0; j < 4; ++j)
#pragma unroll
            for (int rr = 0; rr < 8; ++rr)
                zout[(rowZ + rr + 8 * hi) * DIMZ + 16 * j + lo] = zc[j][rr];

#ifdef HAVE_TDM
        // ---- TDM-prefetch step t+1 into the other buffer, then wait for step t's data ----
        {
            const int tn = (t + 1 < TSTEPS) ? (t + 1) : t;      // clamp: keep 3-in-flight invariant
            const size_t rowN = (size_t)tn * NSIMS + simBase;
            const int p1 = (t + 1) & 1;
            tdm_load_1d((unsigned)((p1 ? NZ1 : NZ0) * 4),   &nzg[rowN * DIMZ], 16 * DIMZ);
            tdm_load_1d((unsigned)((p1 ? NX1 : NX0) * 4),   &nxg[rowN * DIMX], 16 * DIMX);
            tdm_load_1d((unsigned)((p1 ? UOF1 : UOF0) * 4), &u  [rowN * DIMS], 16 * DIMS);
        }
        __builtin_amdgcn_s_wait_tensorcnt(3);                   // step t's 3 loads complete
        const int nzB = (t & 1) ? NZ1 : NZ0;
        const int nxB = (t & 1) ? NX1 : NX0;
        const int uB  = (t & 1) ? UOF1 : UOF0;
#endif

        // ---- u_t fragments in A layout ----
        v2f ua[4];
#pragma unroll
        for (int kk = 0; kk < 4; ++kk)
#ifdef HAVE_TDM
            ua[kk] = *(const v2f*)&lds[uB + lo * DIMS + 4 * kk + 2 * hi];
#else
            ua[kk] = *(const v2f*)&u[(rowZ + lo) * DIMS + 4 * kk + 2 * hi];
#endif

        // ---- relu(z_t) fragments ----
        v2f ra[16];
#pragma unroll
        for (int kk = 0; kk < 16; ++kk) {
            ra[kk].x = fmaxf(za[kk].x, 0.0f);
            ra[kk].y = fmaxf(za[kk].y, 0.0f);
        }

        // ---- x_t = z_t @ B^T + nx * r   (2 N-tiles, K=64) ----
#pragma unroll
        for (int jx = 0; jx < 2; ++jx) {
            v8f acc = {};
#pragma unroll
            for (int kk = 0; kk < 16; ++kk)
                acc = __builtin_amdgcn_wmma_f32_16x16x4_f32(
                    false, za[kk], false, bfr[kk * 2 + jx], (short)0, acc, false, false);
#pragma unroll
            for (int rr = 0; rr < 8; ++rr) {
                size_t row = rowZ + rr + 8 * hi;
#ifdef HAVE_TDM
                float nx = lds[nxB + (rr + 8 * hi) * DIMX + 16 * jx + lo];
#else
                float nx = nxg[row * DIMX + 16 * jx + lo];
#endif
                xout[row * DIMX + 16 * jx + lo] = acc[rr] + nx * rv[jx];
            }
        }

        // ---- mu = z*diag(A) + relu(z)@W_off^T + u@C^T ; z_next = mu + nz*q ----
#pragma unroll
        for (int j = 0; j < 4; ++j) {
            v8f acc = {};
#pragma unroll
            for (int kk = 0; kk < 16; ++kk)        // relu(z) @ W_off^T
                acc = __builtin_amdgcn_wmma_f32_16x16x4_f32(
                    false, ra[kk], false, wfr[kk * 4 + j], (short)0, acc, false, false);
#pragma unroll
            for (int dk = 0; dk < 4; ++dk)         // z * diag(A)  (sparse-diagonal B tile)
                acc = __builtin_amdgcn_wmma_f32_16x16x4_f32(
                    false, za[4 * j + dk], false, dfr[dk * 4 + j], (short)0, acc, false, false);
#pragma unroll
            for (int kk = 0; kk < 4; ++kk)         // u @ C^T
                acc = __builtin_amdgcn_wmma_f32_16x16x4_f32(
                    false, ua[kk], false, cfr[kk * 4 + j], (short)0, acc, false, false);
            // add process noise; keep C/D copy for next step's z store; stash for relayout
#pragma unroll
            for (int rr = 0; rr < 8; ++rr) {
#ifdef HAVE_TDM
                float nz = lds[nzB + (rr + 8 * hi) * DIMZ + 16 * j + lo];
#else
                float nz = nzg[(rowZ + rr + 8 * hi) * DIMZ + 16 * j + lo];
#endif
                float zn = acc[rr] + nz * qv[j];
                zc[j][rr] = zn;
                lds[ZOFF + (rr + 8 * hi) * DIMZ + 16 * j + lo] = zn;
            }
        }

        // ---- relayout z_next: C/D layout in LDS -> A layout registers ----
        asm volatile("s_wait_dscnt 0" ::: "memory");
#pragma unroll
        for (int kk = 0; kk < 16; ++kk)
            za[kk] = *(const v2f*)&lds[ZOFF + lo * DIMZ + 4 * kk + 2 * hi];
    }
}

extern "C" void kernel_launch(void* const* d_in, const int* in_sizes, int n_in,
                              void* d_out, int out_size, void* d_ws, size_t ws_size,
                              hipStream_t stream) {
    const float* u   = (const float*)d_in[0];  // (T, N, 16)
    const float* z0  = (const float*)d_in[1];  // (N, 64)
    const float* nz  = (const float*)d_in[2];  // (T, N, 64)
    const float* nx  = (const float*)d_in[3];  // (T, N, 32)
    const float* AW  = (const float*)d_in[4];  // (64, 64)
    const float* Cm  = (const float*)d_in[5];  // (64, 16)
    const float* Bm  = (const float*)d_in[6];  // (32, 64)
    const float* Q   = (const float*)d_in[7];  // (64,)
    const float* R   = (const float*)d_in[8];  // (32,)

    float* out  = (float*)d_out;
    float* zout = out;                                        // (T, N, 64)
    float* xout = out + (size_t)TSTEPS * NSIMS * DIMZ;        // (T, N, 32)

    dim3 grid(NSIMS / 16);
    dim3 block(32);
    hipLaunchKernelGGL(plrnn_scan_kernel, grid, block, 0, stream,
                       u, z0, nz, nx, AW, Cm, Bm, Q, R, zout, xout);
}